// ImageResampleLayer_53919019434336
// MI455X (gfx1250) — compile-verified
//
#include <hip/hip_runtime.h>
#include <stdint.h>

// Bilinear image resample (grid-sample) for MI455X / gfx1250.
//
// Shapes fixed by the reference: feature [B=8,H=512,W=512,C=16] f32,
// coordinate [B=8,N=65536,2] f32. Output = mask [B,N] ++ value [B,N,C].
//
// Memory-bound gather kernel:
//  - whole 128MB feature fits MI455X's 192MB L2 -> default (RT) loads keep it hot
//  - 34MB of output is write-once -> nontemporal stores so it doesn't evict feature
//  - coordinates staged once per block into LDS via CDNA5 async-to-LDS DMA
//    (ASYNCcnt path), then broadcast to the 4 lanes sharing each point.

#define BLOCK 256
#define PPB   64          // points per block = BLOCK / 4 (4 lanes per point)

constexpr int B_ = 8;
constexpr int H_ = 512;
constexpr int W_ = 512;
constexpr int C_ = 16;
constexpr int N_ = 65536;

typedef float v4f __attribute__((ext_vector_type(4)));

__global__ __launch_bounds__(BLOCK) void ImageResample_gfx1250_kernel(
    const float* __restrict__ feature,
    const float* __restrict__ coord,
    float* __restrict__ mask_out,   // [B*N]
    float* __restrict__ val_out)    // [B*N, C]
{
    __shared__ float scoord[PPB * 2];

    const int tid = threadIdx.x;
    const long long p0 = (long long)blockIdx.x * PPB;

    // ---- Stage this block's 64 coordinate pairs (512B) into LDS via the
    // ---- CDNA5 async global->LDS mover. Waves 0 and 1 are fully active
    // ---- inside this branch (tid 0..63), so EXEC is all-ones for issuers.
    if (tid < PPB) {
        unsigned      lds_off = (unsigned)(uintptr_t)(&scoord[tid * 2]);
        unsigned long long ga = (unsigned long long)(uintptr_t)(coord + (p0 + tid) * 2);
        asm volatile("global_load_async_to_lds_b64 %0, %1, off"
                     :: "v"(lds_off), "v"(ga)
                     : "memory");
#if __has_builtin(__builtin_amdgcn_s_wait_asynccnt)
        __builtin_amdgcn_s_wait_asynccnt(0);
#else
        asm volatile("s_wait_asynccnt 0" ::: "memory");
#endif
    }
    __syncthreads();

    const int       lp = tid >> 2;       // local point within block
    const int       q  = tid & 3;        // channel quad (4 floats each)
    const long long p  = p0 + lp;        // global point index in [0, B*N)
    const int       b  = (int)(p >> 16); // p / N_   (N_ == 65536)

    const float x = scoord[lp * 2 + 0];
    const float y = scoord[lp * 2 + 1];

    const float x0f = floorf(x);
    const float y0f = floorf(y);
    const float fx  = x - x0f;
    const float fy  = y - y0f;

    int x0 = (int)x0f;
    int y0 = (int)y0f;
    x0 = min(max(x0, 0), W_ - 1);
    y0 = min(max(y0, 0), H_ - 1);
    const int x1 = min(x0 + 1, W_ - 1);
    const int y1 = min(y0 + 1, H_ - 1);

    const bool inb = (x >= 0.0f) && (x <= (float)(W_ - 1)) &&
                     (y >= 0.0f) && (y <= (float)(H_ - 1));
    const float m = inb ? 1.0f : 0.0f;

    // feature[b, y, x, c] -> v4f gathers of channel quad q
    const long long imgBase = (long long)b * (long long)(H_ * W_ * C_);
    const int       c0      = q * 4;
    const v4f*      f4      = (const v4f*)feature;

    const long long i00 = (imgBase + (((long long)y0 * W_ + x0) << 4) + c0) >> 2;
    const long long i01 = (imgBase + (((long long)y0 * W_ + x1) << 4) + c0) >> 2;
    const long long i10 = (imgBase + (((long long)y1 * W_ + x0) << 4) + c0) >> 2;
    const long long i11 = (imgBase + (((long long)y1 * W_ + x1) << 4) + c0) >> 2;

    // Four independent B128 gathers; all hit L2 (feature is L2-resident).
    const v4f v00 = f4[i00];
    const v4f v01 = f4[i01];
    const v4f v10 = f4[i10];
    const v4f v11 = f4[i11];

    const float omx = 1.0f - fx;
    const float omy = 1.0f - fy;

    // top = v00*(1-fx) + v01*fx ; bot likewise ; val = (top*(1-fy)+bot*fy)*mask
    const v4f top = v00 * omx + v01 * fx;
    const v4f bot = v10 * omx + v11 * fx;
    const v4f val = (top * omy + bot * fy) * m;

    // Streaming (non-temporal) stores: write-once data must not evict the
    // L2-resident feature tensor. Per-wave value stores are a contiguous
    // 512B burst (lanes 4p+q -> p*64 + q*16 bytes).
    v4f* vdst = (v4f*)(val_out + (p << 4) + c0);
    __builtin_nontemporal_store(val, vdst);

    if (q == 0) {
        __builtin_nontemporal_store(m, mask_out + p);
    }
}

extern "C" void kernel_launch(void* const* d_in, const int* in_sizes, int n_in,
                              void* d_out, int out_size, void* d_ws, size_t ws_size,
                              hipStream_t stream) {
    (void)in_sizes; (void)n_in; (void)d_ws; (void)ws_size; (void)out_size;

    const float* feature = (const float*)d_in[0];
    const float* coord   = (const float*)d_in[1];

    // d_out = [mask (B*N floats)] ++ [value (B*N*C floats)]
    float* mask_out = (float*)d_out;
    float* val_out  = (float*)d_out + (long long)B_ * N_;

    const long long total_threads = (long long)B_ * N_ * 4;   // 4 lanes per point
    const int       blocks        = (int)(total_threads / BLOCK); // 8192

    ImageResample_gfx1250_kernel<<<blocks, BLOCK, 0, stream>>>(
        feature, coord, mask_out, val_out);
}